// DataEmbedding_v2_62697932587160
// MI455X (gfx1250) — compile-verified
//
#include <hip/hip_runtime.h>
#include <hip/hip_bf16.h>

typedef float v2f __attribute__((ext_vector_type(2)));
typedef float v8f __attribute__((ext_vector_type(8)));

#define BB   16
#define SS   2048
#define CIN  64
#define DM   512
#define KDIM 192            // 3 * CIN
#define MT   32             // M tile (rows of (b,t))
#define NT   32             // N tile (d columns)
#define AST  196            // A LDS row stride (192 + 4) -> bank-spread
#define BST  40             // B LDS row stride (32 + 8)  -> half-wave disjoint banks

// ---------- kernel 1: L1 channel norms ----------
__global__ void norms_kernel(const float* __restrict__ x, float* __restrict__ norms) {
    int idx = blockIdx.x * blockDim.x + threadIdx.x;   // flat (b*S + s)
    if (idx >= BB * SS) return;
    const float4* p = (const float4*)(x + (size_t)idx * CIN);
    float s = 0.f;
#pragma unroll
    for (int i = 0; i < CIN / 4; ++i) {
        float4 v = p[i];
        s += fabsf(v.x) + fabsf(v.y) + fabsf(v.z) + fabsf(v.w);
    }
    norms[idx] = s;
}

// ---------- kernel 2: tc backward scan ----------
__global__ void tc_kernel(const float* __restrict__ norms, float* __restrict__ tc) {
    int idx = blockIdx.x * blockDim.x + threadIdx.x;
    if (idx >= BB * SS) return;
    int b = idx / SS, t = idx % SS;
    float thresh = 0.7f * (norms[idx] + 1e-8f);        // (1-ETA)*(norm+EPS)
    const float* nb = norms + (size_t)b * SS;
    float r = 0.f;
    for (int s = t - 1; s >= 0; --s) {                 // largest s with cond -> smallest tau
        if (nb[s] < thresh) { r = (float)(t - s); break; }
    }
    tc[idx] = r;
}

// ---------- kernel 3: fused circular-conv GEMM (fp32 WMMA) + PE + tc embed ----------
__launch_bounds__(128)
__global__ void embed_gemm_kernel(const float* __restrict__ x,
                                  const float* __restrict__ conv_w,
                                  const float* __restrict__ tc_w,
                                  const float* __restrict__ tc_b,
                                  const float* __restrict__ tcv,
                                  float* __restrict__ out) {
    __shared__ float As[MT * AST];
    __shared__ float Bs[KDIM * BST];

    const int tid     = threadIdx.x;
    const int n_block = blockIdx.x * NT;               // d-column base
    const int b       = blockIdx.y >> 6;               // S/MT = 64 row-tiles per batch
    const int t_base  = (blockIdx.y & 63) * MT;

    // Load A tile: 32 rows x 192 features, feature j = k*64 + c -> x[b, (t+k-1) mod S, c]
    const float* xb = x + (size_t)b * SS * CIN;
    for (int i = tid; i < MT * 48; i += 128) {         // 48 float4 groups per row
        int row = i / 48;
        int g   = i % 48;
        int k   = g >> 4;                              // 0..2
        int c4  = (g & 15) * 4;                        // 0..60
        int s   = t_base + row + k - 1;
        if (s < 0)    s += SS;                         // circular wrap
        if (s >= SS)  s -= SS;
        float4 v = *(const float4*)(xb + (size_t)s * CIN + c4);
        float* dst = &As[row * AST + k * CIN + c4];
        dst[0] = v.x; dst[1] = v.y; dst[2] = v.z; dst[3] = v.w;
    }
    // Load B tile: Bs[j][n] = conv_w[(n_block+n)*192 + c*3 + k],  j = k*64 + c
    for (int i = tid; i < KDIM * NT; i += 128) {
        int n = i & 31;
        int j = i >> 5;
        int c = j & 63;
        int k = j >> 6;
        Bs[j * BST + n] = conv_w[(size_t)(n_block + n) * KDIM + c * 3 + k];
    }
    __syncthreads();

    const int wave = tid >> 5;
    const int lane = tid & 31;
    const int half = lane >> 4;                        // 0 | 1 (lanes 16-31)
    const int lidx = lane & 15;
    const int wm   = (wave >> 1) * 16;                 // 2x2 wave layout -> 32x32 block tile
    const int wn   = (wave & 1) * 16;

    // A fragment: lane holds row (wm+lidx), K = kk + 2*half + {0,1}
    const float* arow = &As[(wm + lidx) * AST + 2 * half];
    // B fragment: lane holds col (wn+lidx), K rows kk + 2*half + {0,1}
    const float* bcol = &Bs[(2 * half) * BST + wn + lidx];

    v8f acc = {};
#pragma unroll
    for (int kk = 0; kk < KDIM; kk += 4) {
        v2f a = *(const v2f*)(arow + kk);              // 8B-aligned ds_load_b64
        v2f bf;
        bf.x = bcol[kk * BST];
        bf.y = bcol[kk * BST + BST];
        acc = __builtin_amdgcn_wmma_f32_16x16x4_f32(false, a, false, bf,
                                                    (short)0, acc, false, false);
    }

    // Epilogue: out[b,t,d] = conv + pe[t,d] + tc[b,t]*tc_w[d] + tc_b[d]
    const int   d    = n_block + wn + lidx;            // fixed per lane
    const float w    = tc_w[d];
    const float bias = tc_b[d];
    // div_term = exp(-(d & ~1) * ln(10000)/512)
    const float div  = __expf(-(float)(d & ~1) * 0.017988946039015984f);
    const float* tcb = tcv + (size_t)b * SS;
    float* ob = out + (size_t)b * SS * DM + d;
#pragma unroll
    for (int v = 0; v < 8; ++v) {                      // D-matrix: M = v + 8*half, N = lidx
        int t = t_base + wm + v + 8 * half;
        float ang = (float)t * div;
        float pe  = (d & 1) ? __cosf(ang) : __sinf(ang);
        ob[(size_t)t * DM] = acc[v] + pe + tcb[t] * w + bias;
    }
}

extern "C" void kernel_launch(void* const* d_in, const int* in_sizes, int n_in,
                              void* d_out, int out_size, void* d_ws, size_t ws_size,
                              hipStream_t stream) {
    const float* x      = (const float*)d_in[0];
    const float* conv_w = (const float*)d_in[1];
    const float* tc_w   = (const float*)d_in[2];
    const float* tc_b   = (const float*)d_in[3];
    float* out   = (float*)d_out;
    float* norms = (float*)d_ws;            // B*S floats
    float* tcv   = norms + BB * SS;         // B*S floats

    norms_kernel<<<(BB * SS) / 256, 256, 0, stream>>>(x, norms);
    tc_kernel  <<<(BB * SS) / 256, 256, 0, stream>>>(norms, tcv);

    dim3 grid(DM / NT, (BB * SS) / MT);     // (16, 1024)
    embed_gemm_kernel<<<grid, 128, 0, stream>>>(x, conv_w, tc_w, tc_b, tcv, out);
}